// BinauralLoss_48309792145798
// MI455X (gfx1250) — compile-verified
//
#include <hip/hip_runtime.h>

// ---------------------------------------------------------------------------
// Binaural auditory-model loss for MI455X (gfx1250, wave32).
// Sequential IIR front-end: one thread per filter lane (latency-bound).
// Polyphase resample (the only dense-FLOP stage, ~7 GFLOP) as a banded
// Toeplitz GEMM with V_WMMA_F32_16X16X4_F32; B-operand staged into LDS by
// the Tensor Data Mover (TENSOR_LOAD_TO_LDS + s_wait_tensorcnt).
// Hot loop: 2x ds_load_2addr_b32 + 1x wmma, scalar loop control, EXEC
// untouched (ISA requirement). Output stored transposed -> b128 stores.
// ---------------------------------------------------------------------------

typedef float v2f __attribute__((ext_vector_type(2)));
typedef float v8f __attribute__((ext_vector_type(8)));
typedef unsigned int u32x4 __attribute__((ext_vector_type(4)));
typedef int i32x4 __attribute__((ext_vector_type(4)));
typedef int i32x8 __attribute__((ext_vector_type(8)));

#define PI_F 3.14159265358979f

constexpr int   T      = 32000;   // input samples
constexpr int   TNEW   = 12000;   // resampled samples (ceil(3*T/8))
constexpr int   TNEWP  = 12032;   // padded to 752 full 16-row tiles
constexpr int   NCH    = 29;      // gammatone channels
constexpr int   NFINE  = 17;      // channels with cf < 1400 Hz
constexpr int   LANES  = 704;     // 44 * 16, padded resample-lane count
constexpr int   HC_OFF = 0;       // 232 lanes: hc (ch,sig,b)
constexpr int   FR_OFF = 232;     // 136 lanes: fine real
constexpr int   FI_OFF = 368;     // 136 lanes: fine imag
constexpr int   MR_OFF = 504;     //  96 lanes: mod real
constexpr int   MI_OFF = 600;     //  96 lanes: mod imag
constexpr int   KW     = 354;     // resample taps per phase (2*173+8)
constexpr int   RESW   = 173;     // resample half-width
constexpr int   GUARD  = 320;     // zero guard rows above/below S
constexpr int   SROWS  = T + 2 * GUARD;       // 32640 padded rows
constexpr int   WPAD   = 64;      // weight guard entries each side
constexpr int   WROW   = KW + 2 * WPAD;       // 482 padded taps per phase
constexpr int   NROWS  = 720;     // LDS-staged S rows per block (max span 706)
constexpr float FS16   = 16000.f;
constexpr float FS6    = 6000.f;
constexpr float ADJf   = 0.446683592150963f;  // 10^(8/20)*10^(-15/20)
constexpr float COMPf  = 0.4f;

#if __has_builtin(__builtin_amdgcn_tensor_load_to_lds)
#define HAVE_TDM 1
#else
#define HAVE_TDM 0
#endif

// ---------------- biquad via bilinear transform (prewarped) ----------------
struct BQ {
  float b0, b1, b2, a1, a2, z1, z2;
  __device__ void bilin(float B2, float B1, float B0,
                        float A2, float A1, float A0, float fs) {
    float K  = 2.f * fs;
    float a0 = A2 * K * K + A1 * K + A0;
    b0 = (B2 * K * K + B1 * K + B0) / a0;
    b1 = (2.f * B0 - 2.f * B2 * K * K) / a0;
    b2 = (B2 * K * K - B1 * K + B0) / a0;
    a1 = (2.f * A0 - 2.f * A2 * K * K) / a0;
    a2 = (A2 * K * K - A1 * K + A0) / a0;
    z1 = z2 = 0.f;
  }
  __device__ void lp2(float f, float zeta, float fs) {
    float wc = 2.f * fs * __tanf(PI_F * f / fs);
    bilin(0.f, 0.f, wc * wc, 1.f, 2.f * zeta * wc, wc * wc, fs);
  }
  __device__ void lp1(float f, float fs) {
    float wc = 2.f * fs * __tanf(PI_F * f / fs);
    bilin(0.f, 0.f, wc, 0.f, 1.f, wc, fs);
  }
  __device__ void bp1(float f1, float f2, float fs) {
    float w1 = 2.f * fs * __tanf(PI_F * f1 / fs);
    float w2 = 2.f * fs * __tanf(PI_F * f2 / fs);
    bilin(0.f, w2 - w1, 0.f, 1.f, w2 - w1, w1 * w2, fs);
  }
  __device__ float step(float x) {   // DF2T
    float y = b0 * x + z1;
    z1 = b1 * x - a1 * y + z2;
    z2 = b2 * x - a2 * y;
    return y;
  }
};

// ---------------- complex one-pole gammatone cascade -----------------------
__device__ __forceinline__ float cf_of(int ch) {
  const float LQ = 24.7f * 9.265f;                  // GFB_L * GFB_Q
  float eb = 9.265f * __logf(1.f + 700.f / LQ);     // erb(700)
  float e  = eb + (float)(ch - 11);                 // -n_lo = -11
  return LQ * (__expf(e / 9.265f) - 1.f);
}

struct GT3 {
  float cr, ci, norm;
  float yr[3], yi[3];
  __device__ void initErb(float cf) {
    float erb = 24.7f + cf / 9.265f;
    float a_g = PI_F * 0.375f;
    float lam = __expf(-2.f * PI_F * (erb / a_g) / FS16);
    float ang = 2.f * PI_F * cf / FS16;
    cr = lam * __cosf(ang);
    ci = lam * __sinf(ang);
    float o = 1.f - lam;
    norm = 2.f * o * o * o;
    yr[0] = yr[1] = yr[2] = yi[0] = yi[1] = yi[2] = 0.f;
  }
  __device__ void feed(float xr, float xi, float& outR, float& outI) {
    float ir = xr, ii = xi;
#pragma unroll
    for (int k = 0; k < 3; ++k) {
      float nr = ir + cr * yr[k] - ci * yi[k];
      float ni = ii + cr * yi[k] + ci * yr[k];
      yr[k] = nr; yi[k] = ni; ir = nr; ii = ni;
    }
    outR = ir; outI = ii;
  }
};

// ============ zero the guard rows of padded S ==============================
__global__ void k_pad(float* __restrict__ Spad) {
  int i = blockIdx.x * blockDim.x + threadIdx.x;
  int total = 2 * GUARD * LANES;
  if (i >= total) return;
  int r = i / LANES, c = i - r * LANES;
  int row = (r < GUARD) ? r : (T + r);  // [0,GUARD) and [GUARD+T, SROWS)
  Spad[(size_t)row * LANES + c] = 0.f;
}

__global__ void k_zero(float* out) {
  if (threadIdx.x < 3) out[threadIdx.x] = 0.f;
}

// ================= K0: middle-ear bandpass (8 lanes) =======================
__global__ void k_me(const float* __restrict__ mRef,
                     const float* __restrict__ mTest,
                     float* __restrict__ me) {
  int l = threadIdx.x;
  if (l >= 8) return;
  int s = l >> 1, b = l & 1;
  const float* src = (s == 0 || s == 2) ? mRef : mTest;
  int mc = (s >= 2) ? 1 : 0;
  BQ bq; bq.bp1(500.f, 2000.f, FS16);
  for (int t = 0; t < T; ++t) {
    float x = (t == 0) ? 1e-5f : ADJf * src[(t - 1) * 4 + mc * 2 + b];
    me[t * 8 + l] = bq.step(x);
  }
}

// ============ K1: gammatone + compression + LP chains (464 lanes) ==========
__global__ void k_frontend(const float* __restrict__ me,
                           float* __restrict__ Spad) {
  int l = blockIdx.x * blockDim.x + threadIdx.x;
  if (l >= 464) return;
  float* S = Spad + (size_t)GUARD * LANES;  // row 0 == t 0

  if (l < 232) {                       // ---- hc lanes -> resample input ----
    int ch = l >> 3, sb = l & 7;
    GT3 g; g.initErb(cf_of(ch));
    for (int t = 0; t < T; ++t) {
      float r, i;
      g.feed(me[t * 8 + sb] * g.norm, 0.f, r, i);
      S[(size_t)t * LANES + HC_OFF + l] = __powf(fmaxf(r, 1e-16f), COMPf);
    }
  } else if (l < 368) {                // ---- fine lanes (re+im columns) ----
    int k = l - 232;
    int ch = k >> 3, sb = k & 7;
    float cf = cf_of(ch);
    GT3 g1; g1.initErb(cf);
    GT3 g2; g2.initErb(cf);            // COEF_FINE == COEF_FULL[:NFINE]
    BQ h1, h2, h3;                     // 5th-order Butterworth LP @ 770 Hz
    h1.lp2(770.f, 0.30901699f, FS16);
    h2.lp2(770.f, 0.80901699f, FS16);
    h3.lp1(770.f, FS16);
    for (int t = 0; t < T; ++t) {
      float r, i;
      g1.feed(me[t * 8 + sb] * g1.norm, 0.f, r, i);
      float hc = __powf(fmaxf(r, 1e-16f), COMPf);
      float y  = h3.step(h2.step(h1.step(hc)));
      float fr, fi;
      g2.feed(y * g2.norm, 0.f, fr, fi);
      S[(size_t)t * LANES + FR_OFF + k] = fr;
      S[(size_t)t * LANES + FI_OFF + k] = fi;
    }
  } else {                             // ---- modulation lanes (re+im) -----
    int k = l - 368;
    int mch = k >> 3, sb = k & 7;
    int ch = NFINE + mch;
    GT3 g1; g1.initErb(cf_of(ch));
    BQ h1, h2, h3;
    h1.lp2(770.f, 0.30901699f, FS16);
    h2.lp2(770.f, 0.80901699f, FS16);
    h3.lp1(770.f, FS16);
    BQ m1, m2;
    m1.lp2(500.f, 0.70710678f, FS16);
    m2.lp2(600.f, 0.70710678f, FS16);
    // modulation gammatone (order 1), coef from _gfb_bw(300, 300, 3, 1)
    float phi = PI_F * 300.f / FS16;
    float u   = __powf(10.f, -0.3f);
    float p   = (-2.f + 2.f * u * __cosf(phi)) / (1.f - u);
    float lam = -0.5f * p - __fsqrt_rn(0.25f * p * p - 1.f);
    float beta = 2.f * PI_F * 300.f / FS16;
    float mcr = lam * __cosf(beta), mci = lam * __sinf(beta);
    float mnorm = 2.f * (1.f - lam);
    float prev = 0.f, gyr = 0.f, gyi = 0.f;
    for (int t = 0; t < T; ++t) {
      float r, i;
      g1.feed(me[t * 8 + sb] * g1.norm, 0.f, r, i);
      float hc = __powf(fmaxf(r, 1e-16f), COMPf);
      float y  = h3.step(h2.step(h1.step(hc)));
      float d  = y - prev; prev = y;
      float ml = m2.step(m1.step(d));
      float x  = ml * mnorm;
      float nr = x + mcr * gyr - mci * gyi;
      float ni =     mcr * gyi + mci * gyr;
      gyr = nr; gyi = ni;
      S[(size_t)t * LANES + MR_OFF + k] = gyr;
      S[(size_t)t * LANES + MI_OFF + k] = gyi;
    }
  }
}

// ======== K2: polyphase resample as banded-Toeplitz WMMA GEMM ==============
// Out[q, c] = sum_j Kern[q%3, j] * S[(q/3)*8 + j - 173, c]
// Block: 8 waves = 8 consecutive 16(q) x 16(c) tiles sharing one column set.
// S window (NROWS x 16) DMA'd into LDS by the Tensor Data Mover; weights
// built into LDS with zero guard zones. Fully uniform control flow: scalar
// loop bounds (readfirstlane), no EXEC masking anywhere (WMMA requirement),
// unconditional b128 stores into transposed, padded output.
__global__ void k_resample(const float* __restrict__ Spad,
                           float* __restrict__ Y) {
  __shared__ float wlds[3 * WROW];
  __shared__ float sl[NROWS * 16];

  // ---- polyphase sinc kernel with zero guards ----
  for (int i = threadIdx.x; i < 3 * WROW; i += blockDim.x) {
    int ph = i / WROW, jj = i - ph * WROW - WPAD;
    float v = 0.f;
    if (jj >= 0 && jj < KW) {
      float idx = (float)(jj - RESW) * 0.125f;       // arange(-w, w+L)/L
      float t = (-(float)ph / 3.f + idx) * 2.97f;    // base = min(L,M)*0.99
      t = fminf(fmaxf(t, -64.f), 64.f);
      float tp = t * PI_F;
      float sinc = (fabsf(tp) < 1e-6f) ? 1.f : (__sinf(tp) / tp);
      float c = __cosf(tp * (1.f / 128.f));          // cos^2 window, lpw=64
      v = sinc * c * c * (2.97f / 8.f);
    }
    wlds[i] = v;
  }

  // ---- stage S window into LDS ----
  int qb0 = blockIdx.y * 128;                        // first q of block
  int tb0 = (8 * (qb0 / 3) - RESW) & ~3;             // first staged row
  int c0  = blockIdx.x * 16;
  const float* gsrc = Spad + ((size_t)(tb0 + GUARD) * LANES + c0);

#if HAVE_TDM
  if (threadIdx.x < 32) {                            // wave 0 issues the DMA
    unsigned long long ga = (unsigned long long)(uintptr_t)gsrc;
    unsigned ldsa = (unsigned)(unsigned long long)(uintptr_t)(&sl[0]);
    u32x4 g0 = { 1u,                                 // count=1, user mode
                 ldsa,                               // lds_addr (bytes)
                 (unsigned)(ga & 0xFFFFFFFFu),       // global_addr[31:0]
                 (unsigned)((ga >> 32) & 0x1FFFFFFu) | (2u << 30) }; // +type=2
    i32x8 g1 = { (int)(2u << 16),                    // data_size = 4 bytes
                 (int)(((unsigned)LANES & 0xFFFFu) << 16),   // tensor_dim0 lo16
                 (int)(((unsigned)SROWS & 0xFFFFu) << 16),   // tensor_dim1 lo16
                 (int)(16u << 16),                   // dim1 hi16=0 | tile_dim0=16
                 (int)NROWS,                         // tile_dim1 | tile_dim2=0
                 (int)LANES,                         // tensor_dim0_stride lo32
                 0, 0 };
    i32x4 gz = { 0, 0, 0, 0 };
#if __clang_major__ >= 23
    i32x8 gz8 = { 0, 0, 0, 0, 0, 0, 0, 0 };
    __builtin_amdgcn_tensor_load_to_lds(g0, g1, gz, gz, gz8, 0);
#else
    __builtin_amdgcn_tensor_load_to_lds(g0, g1, gz, gz, 0);
#endif
    __builtin_amdgcn_s_wait_tensorcnt(0);
  }
#else
  for (int i = threadIdx.x; i < NROWS * 16; i += blockDim.x) {
    int r = i >> 4, c = i & 15;
    sl[i] = Spad[(size_t)(tb0 + GUARD + r) * LANES + c0 + c];
  }
#endif
  __syncthreads();

  // ---- per-wave 16x16 output tile (tiles >= TNEW land in Y padding) ----
  int wave = threadIdx.x >> 5;
  int lane = threadIdx.x & 31;
  int q0 = (blockIdx.y * 8 + wave) * 16;

  int m   = lane & 15;                               // A row == B col index
  int kk0 = (lane >> 4) << 1;                        // K slots {0,1} / {2,3}
  int q   = q0 + m;
  int ph  = q % 3;
  int jb  = RESW - 8 * (q / 3);                      // tap j = t + jb

  int tmin = __builtin_amdgcn_readfirstlane((8 * (q0 / 3) - RESW) & ~3);
  int tmax = __builtin_amdgcn_readfirstlane(
      8 * ((q0 + 15) / 3) + (KW - 1) - RESW);
  int niter = ((tmax - tmin) >> 2) + 1;              // scalar trip count

  const float* wli = &wlds[ph * WROW + WPAD + jb + tmin + kk0];
  const float* sbi = &sl[(size_t)(tmin - tb0 + kk0) * 16 + m];

  v8f acc = {};
  for (int it = 0; it < niter; ++it) {
    v2f a, b;
    a.x = wli[0];
    a.y = wli[1];
    b.x = sbi[0];
    b.y = sbi[16];
    acc = __builtin_amdgcn_wmma_f32_16x16x4_f32(false, a, false, b,
                                                (short)0, acc, false, false);
    wli += 4;
    sbi += 64;
  }

  // D layout: VGPR r, lanes 0-15 -> M=r, lanes 16-31 -> M=r+8; N = lane&15.
  // Transposed output Y[col][row]: 8 consecutive floats -> two b128 stores.
  int rowbase = q0 + ((lane >> 4) << 3);
  float* dst = &Y[(size_t)(c0 + m) * TNEWP + rowbase];
  *(float4*)(dst)     = make_float4(acc[0], acc[1], acc[2], acc[3]);
  *(float4*)(dst + 4) = make_float4(acc[4], acc[5], acc[6], acc[7]);
}

// ============ K3: ILD / IVS / IPD smoothing + reduction ====================
// Y is transposed [LANES][TNEWP]: each lane's time series is contiguous.
__global__ void k_post(const float* __restrict__ Y, float* __restrict__ out) {
  int tid = threadIdx.x;

  if (tid < 58) {                          // -------- ILD lanes (ch, b) -----
    int ch = tid >> 1, b = tid & 1;
    BQ lp[4];
#pragma unroll
    for (int s = 0; s < 4; ++s) lp[s].lp2(30.f, 0.70710678f, FS6);
    float acc = 0.f;
    for (int t = 0; t < TNEW; ++t) {
      float v[4];
#pragma unroll
      for (int s = 0; s < 4; ++s)
        v[s] = lp[s].step(Y[(size_t)(HC_OFF + ch * 8 + s * 2 + b) * TNEWP + t]);
      float ildR = 20.f * __log10f(fmaxf(v[2], 1e-4f) / fmaxf(v[0], 1e-4f)) / COMPf;
      float ildT = 20.f * __log10f(fmaxf(v[3], 1e-4f) / fmaxf(v[1], 1e-4f)) / COMPf;
      acc += fabsf(ildR - ildT);
    }
    atomicAdd(&out[2], acc / (float)(TNEW * NCH * 2));
  } else if (tid >= 64 && tid < 64 + 34) { // -------- fine IVS+IPD ----------
    int k = tid - 64;
    int ch = k >> 1, b = k & 1;
    float cf = cf_of(ch);
    float a  = __expf(-cf / (5.f * FS6));
    float Kc = 2.f / (1.f + __expf(0.0012f * (cf - 666.f))) + 0.023f;
    float Nc = 3.17f / (1.f + __expf(-0.0015f * (cf - 560.f))) - 2.75f;
    float scr[2] = {0.f, 0.f}, sci[2] = {0.f, 0.f}, sp[2] = {0.f, 0.f};
    float aIpd = 0.f, aIvs = 0.f;
    for (int t = 0; t < TNEW; ++t) {
      float ipd[2], dv[2];
#pragma unroll
      for (int j = 0; j < 2; ++j) {        // j=0 ref (sig0,2), j=1 test (1,3)
        int lL = ch * 8 + j * 2 + b;
        int lR = ch * 8 + (2 + j) * 2 + b;
        float Lr = Y[(size_t)(FR_OFF + lL) * TNEWP + t];
        float Li = Y[(size_t)(FI_OFF + lL) * TNEWP + t];
        float Rr = Y[(size_t)(FR_OFF + lR) * TNEWP + t];
        float Ri = Y[(size_t)(FI_OFF + lR) * TNEWP + t];
        float ir = Rr * Lr + Ri * Li;      // itf = R * conj(L)
        float ii = Ri * Lr - Rr * Li;
        scr[j] = (1.f - a) * ir + a * scr[j];
        sci[j] = (1.f - a) * ii + a * sci[j];
        sp[j]  = (1.f - a) * __fsqrt_rn(ir * ir + ii * ii) + a * sp[j];
        float ivs = __fsqrt_rn(scr[j] * scr[j] + sci[j] * sci[j]) /
                    fmaxf(sp[j], 1e-16f);
        ipd[j] = atan2f(ii + 1e-16f, ir + 1e-16f);
        dv[j]  = __expf(Kc + Nc) - __expf(Kc * ivs + Nc);
      }
      aIpd += fabsf(ipd[0] - ipd[1]);
      aIvs += fabsf(dv[0] - dv[1]);
    }
    atomicAdd(&out[0], aIpd / (float)(TNEW * NFINE * 2));
    atomicAdd(&out[1], aIvs / (float)(TNEW * NCH * 2));
  } else if (tid >= 104 && tid < 104 + 24) { // ------ modulation IVS --------
    int k = tid - 104;
    int mch = k >> 1, b = k & 1;
    float cf = cf_of(NFINE + mch);
    float a  = __expf(-300.f / (5.f * FS6));
    float Kc = 2.f / (1.f + __expf(0.0012f * (cf - 666.f))) + 0.023f;
    float Nc = 3.17f / (1.f + __expf(-0.0015f * (cf - 560.f))) - 2.75f;
    float scr[2] = {0.f, 0.f}, sci[2] = {0.f, 0.f}, sp[2] = {0.f, 0.f};
    float aIvs = 0.f;
    for (int t = 0; t < TNEW; ++t) {
      float dv[2];
#pragma unroll
      for (int j = 0; j < 2; ++j) {
        int lL = mch * 8 + j * 2 + b;
        int lR = mch * 8 + (2 + j) * 2 + b;
        float Lr = Y[(size_t)(MR_OFF + lL) * TNEWP + t];
        float Li = Y[(size_t)(MI_OFF + lL) * TNEWP + t];
        float Rr = Y[(size_t)(MR_OFF + lR) * TNEWP + t];
        float Ri = Y[(size_t)(MI_OFF + lR) * TNEWP + t];
        float ir = Rr * Lr + Ri * Li;
        float ii = Ri * Lr - Rr * Li;
        scr[j] = (1.f - a) * ir + a * scr[j];
        sci[j] = (1.f - a) * ii + a * sci[j];
        sp[j]  = (1.f - a) * __fsqrt_rn(ir * ir + ii * ii) + a * sp[j];
        float ivs = __fsqrt_rn(scr[j] * scr[j] + sci[j] * sci[j]) /
                    fmaxf(sp[j], 1e-16f);
        dv[j] = __expf(Kc + Nc) - __expf(Kc * ivs + Nc);
      }
      aIvs += fabsf(dv[0] - dv[1]);
    }
    atomicAdd(&out[1], aIvs / (float)(TNEW * NCH * 2));
  }
}

// ===========================================================================
extern "C" void kernel_launch(void* const* d_in, const int* in_sizes, int n_in,
                              void* d_out, int out_size, void* d_ws,
                              size_t ws_size, hipStream_t stream) {
  (void)in_sizes; (void)n_in; (void)out_size; (void)ws_size;
  const float* mRef  = (const float*)d_in[0];   // (T, 2, B=2) f32
  const float* mTest = (const float*)d_in[1];   // (T, 2, B=2) f32
  float* out = (float*)d_out;                   // 3 floats

  char* ws = (char*)d_ws;
  size_t offME = 0;
  size_t offS  = offME + (size_t)T * 8 * sizeof(float);
  size_t offY  = offS  + (size_t)SROWS * LANES * sizeof(float);
  float* me   = (float*)(ws + offME);           //  1.0 MB
  float* Spad = (float*)(ws + offS);            // 91.9 MB (incl. guards)
  float* Yb   = (float*)(ws + offY);            // 33.9 MB, [LANES][TNEWP]

  k_zero<<<1, 32, 0, stream>>>(out);
  k_pad<<<(2 * GUARD * LANES + 255) / 256, 256, 0, stream>>>(Spad);
  k_me<<<1, 32, 0, stream>>>(mRef, mTest, me);
  k_frontend<<<2, 256, 0, stream>>>(me, Spad);
  dim3 grid(LANES / 16, (TNEW / 16 + 7) / 8);   // (44, 94), 8 waves/block
  k_resample<<<grid, 256, 0, stream>>>(Spad, Yb);
  k_post<<<1, 128, 0, stream>>>(Yb, out);
}